// FeatureAttentionLayer_1709396984382
// MI455X (gfx1250) — compile-verified
//
#include <hip/hip_runtime.h>
#include <hip/hip_bf16.h>

// GATv2-style feature attention, fused single kernel for gfx1250 (MI455X).
// Inner pairwise scorer mapped to V_WMMA_F32_16X16X4_F32 rank-4 tiles.
// Round 3: all A/B operands prebuilt outside the WMMA region; g-loop fully
// unrolled -> hot body is a pure WMMA + VALU stream with no loads/waits.

typedef __attribute__((ext_vector_type(2))) float v2f;
typedef __attribute__((ext_vector_type(8))) float v8f;

#define KF    128   // n_features (j / i dimension)
#define EDIM  208   // 2*embed_dim
#define NTILE 13    // EDIM / 16
#define NGRP  8     // KF / 16 j-groups
#define ALPHA 0.2f  // LeakyReLU slope

__global__ __launch_bounds__(256)
void gatv2_feature_attn_kernel(const float* __restrict__ x,      // [32,128]
                               const float* __restrict__ lin_w,  // [208,2]
                               const float* __restrict__ lin_b,  // [208]
                               const float* __restrict__ a,      // [208,1]
                               const float* __restrict__ bias,   // [128,128]
                               float* __restrict__ out)          // [32,128]
{
    __shared__ float s_scores[8][KF];   // per-wave private score row

    const int lane = threadIdx.x & 31;
    const int wave = threadIdx.x >> 5;
    const int row  = blockIdx.x * 8 + wave;   // flat (b,i), exactly 4096 waves
    const int b    = row >> 7;
    const int i    = row & 127;

    const int half = lane >> 4;   // 0: lanes 0-15, 1: lanes 16-31
    const int l15  = lane & 15;

    // ---- Preload B-matrix operands (row-invariant) for all 13 e-tiles ----
    // B(4x16) tile t: K0=w0[e], K1=w1[e], K2=lin_b[e], K3=0
    //   VGPR0: lanes0-15 -> K0, lanes16-31 -> K2
    //   VGPR1: lanes0-15 -> K1, lanes16-31 -> K3
    v2f   Bt[NTILE];
    float av[NTILE];    // a[e] broadcast: lane holds a[t*16 + (lane&15)]
    #pragma unroll
    for (int t = 0; t < NTILE; ++t) {
        const int e = t * 16 + l15;
        const float* p0 = half ? (lin_b + e) : (lin_w + 2 * e);
        float b0 = *p0;                      // branch-free address select
        float w1 = lin_w[2 * e + 1];         // unconditional, always in-range
        Bt[t].x = b0;
        Bt[t].y = half ? 0.0f : w1;
        av[t]   = a[e];
    }

    // ---- Preload the whole x[b,:] row: lane holds x[b, lane + 32k] ----
    float xr[4];
    #pragma unroll
    for (int k = 0; k < 4; ++k)
        xr[k] = x[b * KF + lane + 32 * k];

    const float u  = x[b * KF + i];
    const float Ax = half ? 1.0f : u;        // A VGPR0: K0=u / K2=1.0

    // ---- Prebuild all 8 A-matrix operands (hoists every ds_bpermute /
    //      cndmask out of the WMMA region). A(16x4) row j: (u, v_j, 1, 0). ----
    v2f Ag[NGRP];
    #pragma unroll
    for (int g = 0; g < NGRP; ++g) {
        // vj = x[b, g*16 + l15]; g is a constant here -> fixed source register
        const float vj = __shfl(xr[g >> 1], ((g & 1) << 4) + l15, 32);
        Ag[g].x = Ax;                        // K0=u / K2=1.0
        Ag[g].y = half ? 0.0f : vj;          // K1=v_j / K3=0
    }

    // ---- Scores: e[i,j] = sum_e a[e]*LReLU(u*w0 + v_j*w1 + b) ----
    #pragma unroll
    for (int g = 0; g < NGRP; ++g) {         // fully unrolled: 104 WMMAs
        v8f facc = {0.f,0.f,0.f,0.f,0.f,0.f,0.f,0.f};
        #pragma unroll
        for (int t = 0; t < NTILE; ++t) {
            v8f c = {0.f,0.f,0.f,0.f,0.f,0.f,0.f,0.f};
            // h tile: rows M=j (16), cols N=e (16), K=4 rank-4 outer product
            v8f h = __builtin_amdgcn_wmma_f32_16x16x4_f32(
                        false, Ag[g], false, Bt[t], (short)0, c, false, false);
            #pragma unroll
            for (int r = 0; r < 8; ++r) {
                float hv = h[r];
                float lr = fmaxf(hv, ALPHA * hv);        // LeakyReLU
                facc[r]  = __builtin_fmaf(lr, av[t], facc[r]);
            }
        }
        // Reduce over N (=e) : sum across each 16-lane half.
        #pragma unroll
        for (int r = 0; r < 8; ++r) {
            float red = facc[r];
            red += __shfl_xor(red, 1, 16);
            red += __shfl_xor(red, 2, 16);
            red += __shfl_xor(red, 4, 16);
            red += __shfl_xor(red, 8, 16);
            // half0 holds score j = g*16+r ; half1 holds j = g*16+r+8
            if (l15 == 0)
                s_scores[wave][g * 16 + r + half * 8] = red;
        }
    }

    __syncthreads();   // uniform across all 8 waves; orders LDS w->r

    // ---- Softmax over j (128) + out = sum_j p_j * x[b,j], then sigmoid ----
    float s[4];
    float m = -3.4e38f;
    #pragma unroll
    for (int k = 0; k < 4; ++k) {
        const int j = lane + 32 * k;
        s[k] = s_scores[wave][j] + bias[i * KF + j];
        m = fmaxf(m, s[k]);
    }
    #pragma unroll
    for (int off = 16; off >= 1; off >>= 1)
        m = fmaxf(m, __shfl_xor(m, off, 32));

    float sum = 0.0f, dot = 0.0f;
    #pragma unroll
    for (int k = 0; k < 4; ++k) {
        float ex = __expf(s[k] - m);
        sum += ex;
        dot = __builtin_fmaf(ex, xr[k], dot);   // reuse preloaded x row
    }
    #pragma unroll
    for (int off = 16; off >= 1; off >>= 1) {
        sum += __shfl_xor(sum, off, 32);
        dot += __shfl_xor(dot, off, 32);
    }

    const float z = dot / sum;
    const float o = 1.0f / (1.0f + __expf(-z));
    if (lane == 0)
        out[row] = o;
}

extern "C" void kernel_launch(void* const* d_in, const int* in_sizes, int n_in,
                              void* d_out, int out_size, void* d_ws, size_t ws_size,
                              hipStream_t stream) {
    (void)in_sizes; (void)n_in; (void)d_ws; (void)ws_size; (void)out_size;
    const float* x     = (const float*)d_in[0];
    const float* lin_w = (const float*)d_in[1];
    const float* lin_b = (const float*)d_in[2];
    const float* a     = (const float*)d_in[3];
    const float* bias  = (const float*)d_in[4];
    float* out = (float*)d_out;

    // 4096 rows (b,i), one wave32 per row, 8 waves per block.
    dim3 grid(512), block(256);
    hipLaunchKernelGGL(gatv2_feature_attn_kernel, grid, block, 0, stream,
                       x, lin_w, lin_b, a, bias, out);
}